// MeanAggregator_5368709120505
// MI455X (gfx1250) — compile-verified
//
#include <hip/hip_runtime.h>
#include <hip/hip_bf16.h>

typedef __attribute__((ext_vector_type(2))) float v2f;
typedef __attribute__((ext_vector_type(8))) float v8f;

#define KMAX 10
#define ROWF 136              // LDS row stride in floats: 128 + 8 pad (bank-conflict dodge)
#define WAVES_PER_BLOCK 8

__global__ __launch_bounds__(256) void mean_agg_wmma(
    const float* __restrict__ features,
    const int*   __restrict__ neigh_idx,
    const float* __restrict__ neigh_mask,
    float* __restrict__ out,
    int n_nodes)
{
    __shared__ __align__(16) float smem[WAVES_PER_BLOCK * KMAX * ROWF];

    const int lane = threadIdx.x & 31;
    const int wave = threadIdx.x >> 5;
    const int node = blockIdx.x * WAVES_PER_BLOCK + wave;
    if (node >= n_nodes) return;          // wave-uniform; active waves keep EXEC all-1s

    float* sw = &smem[wave * (KMAX * ROWF)];
    unsigned lds_base = (unsigned)(unsigned long long)(void*)sw;  // low 32 bits = LDS offset

    // Lane k (k<10) owns neighbor slot k; others carry weight 0.
    int   kk = (lane < KMAX) ? lane : (KMAX - 1);
    float m  = neigh_mask[(size_t)node * KMAX + kk];
    int   r  = neigh_idx [(size_t)node * KMAX + kk];
    if (lane >= KMAX) m = 0.0f;

    // num_neigh via wave32 butterfly reduction; fold 1/num into the weights.
    float num = m;
    #pragma unroll
    for (int off = 16; off >= 1; off >>= 1) num += __shfl_xor(num, off, 32);
    float ws = m * (1.0f / num);          // per-slot weight (0 for pad slots)

    // Stage the 10 neighbor rows into LDS with CDNA5 async copies (ASYNCcnt path).
    // Each instruction moves one 512B row: lane j copies bytes [16j,16j+16).
    #pragma unroll
    for (int k = 0; k < KMAX; ++k) {
        int rk = __shfl(r, k, 32);        // wave-uniform row index
        unsigned long long ga =
            (unsigned long long)(features + (size_t)rk * 128 + lane * 4);
        unsigned la = lds_base + (unsigned)(k * (ROWF * 4) + lane * 16);
        asm volatile("global_load_async_to_lds_b128 %0, %1, off"
                     :: "v"(la), "v"(ga) : "memory");
    }
    asm volatile("s_wait_asynccnt 0x0" ::: "memory");

    // Accumulators: 8 column tiles of 16 (D = 16x16 f32 -> v8f each).
    v8f acc[8];
    #pragma unroll
    for (int t = 0; t < 8; ++t) {
        v8f z = {0.f, 0.f, 0.f, 0.f, 0.f, 0.f, 0.f, 0.f};
        acc[t] = z;
    }

    const int col = lane & 15;
    // 3 K-chunks of 4 (K=10 padded with zero weights), 8 N-tiles of 16.
    // A layout (ISA 7.12.2): lanes 0-15 -> K0 (vgpr0), K1 (vgpr1); lanes 16-31 -> K2, K3.
    // A rows all equal => every D row equals the weighted sum (exact f32 math).
    #pragma unroll
    for (int c = 0; c < 3; ++c) {
        int kb = 4 * c + ((lane < 16) ? 0 : 2);
        float a0 = __shfl(ws, kb,     32);
        float a1 = __shfl(ws, kb + 1, 32);
        v2f A; A[0] = a0; A[1] = a1;
        int kr0 = (kb     < KMAX) ? kb     : 0;   // pad slots: weight 0, finite data
        int kr1 = (kb + 1 < KMAX) ? kb + 1 : 0;
        const float* r0 = sw + kr0 * ROWF;
        const float* r1 = sw + kr1 * ROWF;
        #pragma unroll
        for (int t = 0; t < 8; ++t) {
            v2f B; B[0] = r0[t * 16 + col]; B[1] = r1[t * 16 + col];
            acc[t] = __builtin_amdgcn_wmma_f32_16x16x4_f32(
                false, A, false, B, (short)0, acc[t], false, false);
        }
    }

    // Store: all D rows are identical; VGPR0 of each tile holds N = lane%16.
    // Lanes 0-15 write tiles 0-3, lanes 16-31 write tiles 4-7 (two 64B segments/store).
    float* orow = out + (size_t)node * 128;
    int half = lane >> 4;
    #pragma unroll
    for (int j = 0; j < 4; ++j) {
        float v = half ? acc[4 + j][0] : acc[j][0];
        orow[half * 64 + j * 16 + col] = v;
    }
}

extern "C" void kernel_launch(void* const* d_in, const int* in_sizes, int n_in,
                              void* d_out, int out_size, void* d_ws, size_t ws_size,
                              hipStream_t stream) {
    const float* features   = (const float*)d_in[0];
    const int*   neigh_idx  = (const int*)  d_in[1];
    const float* neigh_mask = (const float*)d_in[2];
    float* out = (float*)d_out;

    int N = in_sizes[1] / KMAX;          // neigh_idx is [N, 10]
    int blocks = (N + WAVES_PER_BLOCK - 1) / WAVES_PER_BLOCK;
    hipLaunchKernelGGL(mean_agg_wmma, dim3(blocks), dim3(256), 0, stream,
                       features, neigh_idx, neigh_mask, out, N);
}